// ChemConv_31903017074942
// MI455X (gfx1250) — compile-verified
//
#include <hip/hip_runtime.h>
#include <hip/hip_bf16.h>

// ---------------------------------------------------------------------------
// ChemConv on MI455X (gfx1250, wave32, WMMA + async-to-LDS).
//
// out[a,o] = sum_{n,f} conn[a, n*12+f] * Y[n*12+f, o] + bond term
//   Y[n*12+f, o] = sum_i x[n,i] * WcT[i, f*64+o],  WcT = P[o,f,i]*bf[o,f,0]
//
// conn is (A, A, 12) f32 row-major -> row a is contiguous length 24576.
// Main GEMM: M=2048, N=64, K=24576, f32 via V_WMMA_F32_16X16X4_F32,
// double-buffered LDS staged with GLOBAL_LOAD_ASYNC_TO_LDS_B128.
// ---------------------------------------------------------------------------

#define NA    2048
#define IND   64
#define OUTD  64
#define FL    12
#define KTOT  (NA * FL)          // 24576
#define KSPLIT 16
#define KCHUNK (KTOT / KSPLIT)   // 1536
#define KC    64                 // K tile staged in LDS per chunk
#define NCHUNKS (KCHUNK / KC)    // 24
#define NCOL  (FL * OUTD)        // 768
#define MB    128                // rows per workgroup in main GEMM

static constexpr int LDA = KC + 4;    // 68 floats/row: 16B-aligned, conflict-free
static constexpr int LDB = OUTD + 8;  // 72 floats/row: 16B-aligned, conflict-free
#define ASYNC_OPS 12                  // async b128 issues per thread per chunk

typedef __attribute__((ext_vector_type(2))) float v2f;
typedef __attribute__((ext_vector_type(8))) float v8f;

// Builtin-exact vector/pointer types for async copies (per hipcc diagnostic:
// parameter is a pointer to int __attribute__((vector_size(16)))).
typedef int v4i_t __attribute__((vector_size(16)));
typedef v4i_t __attribute__((address_space(1)))* as1_v4i;
typedef v4i_t __attribute__((address_space(3)))* as3_v4i;

#if defined(__has_builtin)
#if __has_builtin(__builtin_amdgcn_global_load_async_to_lds_b128) && \
    __has_builtin(__builtin_amdgcn_s_wait_asynccnt)
#define USE_ASYNC 1
#endif
#endif
#ifndef USE_ASYNC
#define USE_ASYNC 0
#endif

__device__ __forceinline__ v8f wmma4(v2f a, v2f b, v8f c) {
  // D = A(16x4, f32) * B(4x16, f32) + C(16x16, f32)
  return __builtin_amdgcn_wmma_f32_16x16x4_f32(
      /*neg_a=*/false, a, /*neg_b=*/false, b,
      /*c_mod=*/(short)0, c, /*reuse_a=*/false, /*reuse_b=*/false);
}

__device__ __forceinline__ void cp16(float* l, const float* g) {
#if USE_ASYNC
  // AS1 ptrs are 64-bit (same representation as generic global); AS3 ptrs are
  // 32-bit and generic-LDS addr[31:0] is exactly the LDS byte offset.
  __builtin_amdgcn_global_load_async_to_lds_b128(
      (as1_v4i)(uintptr_t)g, (as3_v4i)(uintptr_t)l, 0, 0);
#else
  *(float4*)l = *(const float4*)g;
#endif
}

__device__ __forceinline__ void wait_async_keep() {  // <= ASYNC_OPS in flight
#if USE_ASYNC
  __builtin_amdgcn_s_wait_asynccnt(ASYNC_OPS);
#endif
}
__device__ __forceinline__ void wait_async_all() {
#if USE_ASYNC
  __builtin_amdgcn_s_wait_asynccnt(0);
#endif
}

// --------------------------------------------------------------------------
// 1) WcT[i*768 + f*64 + o] = P[(o*12+f)*64 + i] * BF[(o*12+f)*3 + 0]
// --------------------------------------------------------------------------
__global__ void __launch_bounds__(256)
chem_prepw(const float* __restrict__ P, const float* __restrict__ BF,
           float* __restrict__ WcT) {
  int tid = blockIdx.x * 256 + threadIdx.x;   // 0 .. 49151
  if (tid >= IND * NCOL) return;
  int i   = tid / NCOL;
  int col = tid - i * NCOL;
  int f = col >> 6, o = col & 63;
  WcT[tid] = P[(o * FL + f) * IND + i] * BF[(o * FL + f) * 3 + 0];
}

// --------------------------------------------------------------------------
// 2) Y GEMM: M=2048 (n), N=768 (f*64+o), K=64 (i).
//    Y stored at [((n*12+f)<<6) + o] to match the main GEMM's K layout.
//    grid (128, 6), block 256 (8 waves); wave handles one 16x16 tile.
// --------------------------------------------------------------------------
__global__ void __launch_bounds__(256)
chem_y(const float* __restrict__ X, const float* __restrict__ WcT,
       float* __restrict__ Y) {
  int lane = threadIdx.x & 31;
  int w    = threadIdx.x >> 5;
  int lid  = lane & 15;
  int half = lane >> 4;
  int m0 = blockIdx.x * 16;
  int n0 = (blockIdx.y * 8 + w) * 16;

  v8f acc = {};
#pragma unroll
  for (int kk = 0; kk < IND / 4; ++kk) {
    int k = kk * 4 + 2 * half;
    v2f a, b;
    a.x = X[(m0 + lid) * IND + k];
    a.y = X[(m0 + lid) * IND + k + 1];
    b.x = WcT[k * NCOL + n0 + lid];
    b.y = WcT[(k + 1) * NCOL + n0 + lid];
    acc = wmma4(a, b, acc);
  }
  int f  = n0 >> 6;
  int ob = n0 & 63;
#pragma unroll
  for (int r = 0; r < 8; ++r) {
    int n = m0 + r + 8 * half;
    Y[((n * FL + f) << 6) + ob + lid] = acc[r];
  }
}

// --------------------------------------------------------------------------
// 3) Main split-K GEMM -> part[split][a][o].
//    grid (16, KSPLIT), block 256 = 8 waves.
//    Workgroup tile: 128(M) x 64(N); wave w owns rows [w*16, w*16+16) x 64.
//    conn & Y double-buffered in LDS via async-to-LDS copies.
// --------------------------------------------------------------------------
__device__ __forceinline__ void issue_chunk(const float* __restrict__ C2,
                                            const float* __restrict__ Y,
                                            float* AsB, float* BsB,
                                            int m0, int kb, int t) {
  // conn tile: 128 rows x 64 cols = 2048 float4 (8 per thread, coalesced)
#pragma unroll
  for (int j = 0; j < 8; ++j) {
    int idx = t + j * 256;
    int r = idx >> 4, c4 = (idx & 15) << 2;
    cp16(&AsB[r * LDA + c4], &C2[(size_t)(m0 + r) * KTOT + kb + c4]);
  }
  // Y tile: 64 rows x 64 cols = 1024 float4 (4 per thread, coalesced)
#pragma unroll
  for (int j = 0; j < 4; ++j) {
    int idx = t + j * 256;
    int r = idx >> 4, c4 = (idx & 15) << 2;
    cp16(&BsB[r * LDB + c4], &Y[(size_t)(kb + r) * OUTD + c4]);
  }
}

__global__ void __launch_bounds__(256)
chem_gemm(const float* __restrict__ C2, const float* __restrict__ Y,
          float* __restrict__ part) {
  __shared__ float As[2][MB * LDA];   // 2 * 128*68*4 = 69.6 KB
  __shared__ float Bs[2][KC * LDB];   // 2 * 64*72*4  = 36.9 KB

  int t    = threadIdx.x;
  int lane = t & 31;
  int w    = t >> 5;                  // wave 0..7 -> M-strip
  int lid  = lane & 15;
  int half = lane >> 4;
  int m0  = blockIdx.x * MB;
  int ks0 = blockIdx.y * KCHUNK;

  v8f acc[4] = {v8f{}, v8f{}, v8f{}, v8f{}};

  issue_chunk(C2, Y, As[0], Bs[0], m0, ks0, t);

  for (int ch = 0; ch < NCHUNKS - 1; ++ch) {
    int cur = ch & 1;
    issue_chunk(C2, Y, As[1 - cur], Bs[1 - cur], m0, ks0 + (ch + 1) * KC, t);
    wait_async_keep();               // chunk ch resident, next chunk in flight
    __syncthreads();
    const float* Asc = As[cur];
    const float* Bsc = Bs[cur];
#pragma unroll
    for (int kk = 0; kk < KC / 4; ++kk) {
      int kc = kk * 4 + 2 * half;
      v2f a = *(const v2f*)&Asc[(w * 16 + lid) * LDA + kc];
#pragma unroll
      for (int nt = 0; nt < 4; ++nt) {
        v2f b;
        b.x = Bsc[kc * LDB + nt * 16 + lid];
        b.y = Bsc[(kc + 1) * LDB + nt * 16 + lid];
        acc[nt] = wmma4(a, b, acc[nt]);
      }
    }
    __syncthreads();
  }
  // peeled last chunk
  {
    int cur = (NCHUNKS - 1) & 1;
    wait_async_all();
    __syncthreads();
    const float* Asc = As[cur];
    const float* Bsc = Bs[cur];
#pragma unroll
    for (int kk = 0; kk < KC / 4; ++kk) {
      int kc = kk * 4 + 2 * half;
      v2f a = *(const v2f*)&Asc[(w * 16 + lid) * LDA + kc];
#pragma unroll
      for (int nt = 0; nt < 4; ++nt) {
        v2f b;
        b.x = Bsc[kc * LDB + nt * 16 + lid];
        b.y = Bsc[(kc + 1) * LDB + nt * 16 + lid];
        acc[nt] = wmma4(a, b, acc[nt]);
      }
    }
  }

  float* p = part + (size_t)(blockIdx.y * NA + m0 + w * 16) * OUTD;
#pragma unroll
  for (int nt = 0; nt < 4; ++nt)
#pragma unroll
    for (int r = 0; r < 8; ++r)
      p[(size_t)(r + 8 * half) * OUTD + nt * 16 + lid] = acc[nt][r];
}

// --------------------------------------------------------------------------
// 4) Reduce split-K partials + bond term -> out  (deterministic order)
// --------------------------------------------------------------------------
__global__ void __launch_bounds__(256)
chem_reduce(const float* __restrict__ part, const float* __restrict__ BP,
            const float* __restrict__ BF, float* __restrict__ out) {
  int tid = blockIdx.x * 256 + threadIdx.x;   // 0 .. 2048*64-1
  if (tid >= NA * OUTD) return;
  int a = tid >> 6, o = tid & 63;
  float s = 0.f;
#pragma unroll
  for (int sp = 0; sp < KSPLIT; ++sp)
    s += part[((size_t)(sp * NA + a) << 6) + o];
  float bs = 0.f;
#pragma unroll
  for (int f = 0; f < FL; ++f) {
    bs += BP[(a * FL + f) * 2 + 0] * BF[(o * FL + f) * 3 + 1];
    bs += BP[(a * FL + f) * 2 + 1] * BF[(o * FL + f) * 3 + 2];
  }
  out[tid] = s + bs;
}

// --------------------------------------------------------------------------
extern "C" void kernel_launch(void* const* d_in, const int* in_sizes, int n_in,
                              void* d_out, int out_size, void* d_ws, size_t ws_size,
                              hipStream_t stream) {
  const float* x    = (const float*)d_in[0];   // (2048, 64)
  const float* conn = (const float*)d_in[1];   // (2048, 2048, 12) -> (2048, 24576)
  const float* bp   = (const float*)d_in[2];   // (2048, 12, 2)
  const float* pf   = (const float*)d_in[3];   // (64, 12, 64)
  const float* bf   = (const float*)d_in[4];   // (64, 12, 3)
  float* out = (float*)d_out;                  // (2048, 64)

  float* Y    = (float*)d_ws;                  // 2048*768    = 1,572,864 f
  float* WcT  = Y + (size_t)NA * FL * OUTD;    // 64*768      =    49,152 f
  float* part = WcT + (size_t)IND * NCOL;      // 16*2048*64  = 2,097,152 f

  chem_prepw<<<(IND * NCOL + 255) / 256, 256, 0, stream>>>(pf, bf, WcT);
  chem_y<<<dim3(NA / 16, NCOL / 128), 256, 0, stream>>>(x, WcT, Y);
  chem_gemm<<<dim3(NA / MB, KSPLIT), 256, 0, stream>>>(conn, Y, part);
  chem_reduce<<<(NA * OUTD + 255) / 256, 256, 0, stream>>>(part, bp, bf, out);
}